// build_cost_volume_52802327937445
// MI455X (gfx1250) — compile-verified
//
#include <hip/hip_runtime.h>

// Cost-volume: out[b,c,d,h,w] = (L[b,c,h,w] - R[b,c,h,w-d]) * (w >= d)
// Shapes fixed by the reference: B=8, C=32, H=68, W=120, DISP=24.
// Pure store-bandwidth-bound: ~200 MB of fp32 output, ~9.3 us floor @ 23.3 TB/s.

#define DISP 24
#define Wd   120
#define Hd   68
#define NBC  256      // B*C = 8*32
#define THREADS 256   // 8 wave32 waves per block

typedef float v4f __attribute__((ext_vector_type(4)));
typedef int   v4i __attribute__((ext_vector_type(4)));

#if defined(__gfx1250__) && __has_builtin(__builtin_amdgcn_global_load_async_to_lds_b128)
#define HAVE_ASYNC_LDS 1
#else
#define HAVE_ASYNC_LDS 0
#endif

__global__ __launch_bounds__(THREADS)
void build_cost_volume_52802327937445_kernel(const float* __restrict__ L,
                                             const float* __restrict__ R,
                                             float* __restrict__ out)
{
    __shared__ __align__(16) float sL[Wd];
    __shared__ __align__(16) float sR[Wd];

    const int tid  = threadIdx.x;
    const int row  = blockIdx.x;      // 0 .. NBC*H - 1  == (bc*H + h)
    const int h    = row % Hd;
    const int bc   = row / Hd;
    const int in_off = row * Wd;      // flat offset of this (b,c,h) row

#if HAVE_ASYNC_LDS
    // gfx1250 async global->LDS copy (ASYNCcnt): lanes 0..29 of wave 0 stage
    // the L row, lanes 0..29 of wave 1 stage the R row, 16B per lane.
    {
        const int wv = tid >> 5;
        const int ln = tid & 31;
        if (wv < 2 && ln < 30) {
            const float* gsrc = (wv == 0 ? L : R) + in_off + ln * 4;
            float*       ldst = (wv == 0 ? sL : sR) + ln * 4;
            __builtin_amdgcn_global_load_async_to_lds_b128(
                (__attribute__((address_space(1))) v4i*)(v4i*)(float*)gsrc,
                (__attribute__((address_space(3))) v4i*)(v4i*)ldst,
                0, 0);
        }
    }
#if __has_builtin(__builtin_amdgcn_s_wait_asynccnt)
    __builtin_amdgcn_s_wait_asynccnt(0);
#else
    asm volatile("s_wait_asynccnt 0" ::: "memory");
#endif
    __syncthreads();
#else
    // Fallback staging: global load -> ds_store
    for (int i = tid; i < Wd; i += THREADS) {
        sL[i] = L[in_off + i];
        sR[i] = R[in_off + i];
    }
    __syncthreads();
#endif

    // 24 disparities x 30 float4 = 720 vec4 stores per (b,c,h) row.
    // Consecutive lanes -> consecutive w0 within one disparity row: each
    // wave's 32x16B stores form at most two contiguous runs (30 float4 per row), streamed with NT.
    for (int idx = tid; idx < DISP * (Wd / 4); idx += THREADS) {
        const int d  = idx / (Wd / 4);
        const int w0 = (idx - d * (Wd / 4)) * 4;

        const v4f l = *(const v4f*)(sL + w0);
        const int base = w0 - d;
        // Clamp LDS index for invalid lanes (read sR[0] harmlessly), then select 0.
        const float r0 = sR[base     > 0 ? base     : 0];
        const float r1 = sR[base + 1 > 0 ? base + 1 : 0];
        const float r2 = sR[base + 2 > 0 ? base + 2 : 0];
        const float r3 = sR[base + 3 > 0 ? base + 3 : 0];

        v4f o;
        o.x = (base     >= 0) ? l.x - r0 : 0.0f;
        o.y = (base + 1 >= 0) ? l.y - r1 : 0.0f;
        o.z = (base + 2 >= 0) ? l.z - r2 : 0.0f;
        o.w = (base + 3 >= 0) ? l.w - r3 : 0.0f;

        float* dst = out + ((long)(bc * DISP + d) * Hd + h) * Wd + w0;
        __builtin_nontemporal_store(o, (v4f*)dst);
    }
}

extern "C" void kernel_launch(void* const* d_in, const int* in_sizes, int n_in,
                              void* d_out, int out_size, void* d_ws, size_t ws_size,
                              hipStream_t stream) {
    (void)in_sizes; (void)n_in; (void)out_size; (void)d_ws; (void)ws_size;
    const float* L = (const float*)d_in[0];
    const float* R = (const float*)d_in[1];
    float* out = (float*)d_out;

    dim3 grid(NBC * Hd);   // 17,408 blocks, one per (b,c,h) row
    dim3 block(THREADS);
    hipLaunchKernelGGL(build_cost_volume_52802327937445_kernel,
                       grid, block, 0, stream, L, R, out);
}